// MEGATConv_90331752169544
// MI455X (gfx1250) — compile-verified
//
#include <hip/hip_runtime.h>
#include <math.h>

typedef float v2f __attribute__((ext_vector_type(2)));
typedef float v8f __attribute__((ext_vector_type(8)));
typedef int   v4i __attribute__((ext_vector_type(4)));

#define B_ 4
#define N_ 512
#define K_ 64   // IN_FEATS = H*F
#define H_ 8
#define F_ 8

#if __has_builtin(__builtin_amdgcn_global_load_async_to_lds_b128)
#define HAVE_ASYNC_LDS 1
typedef __attribute__((address_space(1))) v4i gv4i;   // global int4
typedef __attribute__((address_space(3))) v4i lv4i;   // LDS int4
#else
#define HAVE_ASYNC_LDS 0
#endif

__device__ __forceinline__ void wait_async0() {
#if __has_builtin(__builtin_amdgcn_s_wait_asynccnt)
  __builtin_amdgcn_s_wait_asynccnt(0);
#else
  asm volatile("s_wait_asynccnt 0x0" ::: "memory");
#endif
}

// -------- helpers --------
__device__ __forceinline__ v8f wmma4(v2f a, v2f b, v8f c) {
  // D = A(16x4 f32) * B(4x16 f32) + C(16x16 f32)
  return __builtin_amdgcn_wmma_f32_16x16x4_f32(false, a, false, b, (short)0, c,
                                               false, false);
}
__device__ __forceinline__ float elu1(float x) { return x > 0.f ? x : expm1f(x); }
__device__ __forceinline__ float lrelu(float x) { return x > 0.f ? x : 0.2f * x; }

// ======================================================================
// Kernel 1: xp = x @ W  (per-row),  s_src/s_dst head reductions
// ======================================================================
__global__ void proj_nodes_kernel(const float* __restrict__ x,
                                  const float* __restrict__ W,
                                  const float* __restrict__ a_src,
                                  const float* __restrict__ a_dst,
                                  float* __restrict__ xp,
                                  float* __restrict__ ssrc,
                                  float* __restrict__ sdst) {
  __shared__ float xr[K_];
  const int row = blockIdx.x;          // b*512 + n
  const int t = threadIdx.x;           // 0..63
  xr[t] = x[(size_t)row * K_ + t];
  __syncthreads();
  float acc = 0.f;
#pragma unroll
  for (int k = 0; k < K_; ++k) acc += xr[k] * W[k * K_ + t];
  xp[(size_t)row * K_ + t] = acc;
  float ps = acc * a_src[t];
  float pd = acc * a_dst[t];
  ps += __shfl_xor(ps, 1); ps += __shfl_xor(ps, 2); ps += __shfl_xor(ps, 4);
  pd += __shfl_xor(pd, 1); pd += __shfl_xor(pd, 2); pd += __shfl_xor(pd, 4);
  if ((t & 7) == 0) {
    const int b = row >> 9, n = row & 511, h = t >> 3;
    ssrc[((b * H_ + h) << 9) + n] = ps;
    sdst[((b * H_ + h) << 9) + n] = pd;
  }
}

// ======================================================================
// Kernel 2 (dominant): per (b,i) WG.
//   ep[j,:] = e[b,i,j,:] @ We  via v_wmma_f32_16x16x4_f32
//   e_out = elu(ep)  (non-temporal; 268 MB stream)
//   sedge[b,h,j,i] = sum_f ep[j,h*8+f]*a_edge[h,f]
// Loop order: nt outer (B frags held in 32 VGPRs), jt inner -> no spills.
// e tile staged via GLOBAL_LOAD_ASYNC_TO_LDS_B128 when available.
// ======================================================================
__global__ void edge_kernel(const float* __restrict__ e,
                            const float* __restrict__ We,
                            const float* __restrict__ a_edge,
                            float* __restrict__ e_out,
                            float* __restrict__ sedge) {
  extern __shared__ float smem[];
  float* et = smem;                  // 512 * 68
  float* wt = smem + 512 * 68;       // 64 * 68  (WeT[n][k])
  float* ae = wt + 64 * 68;          // 64
  const int tid = threadIdx.x, lane = tid & 31, wave = tid >> 5;
  const int b = blockIdx.x >> 9, i = blockIdx.x & 511;
  const size_t ebase = (size_t)(b * N_ + i) * (N_ * K_);

  // ---- stage e[b,i,:,:] : contiguous 128 KB ----
  const float4* esrc = (const float4*)(e + ebase);
#pragma unroll
  for (int it = 0; it < 32; ++it) {
    const int idx4 = tid + it * 256;           // 8192 float4 total
    const int j = idx4 >> 4, kq = (idx4 & 15) << 2;
#if HAVE_ASYNC_LDS
    __builtin_amdgcn_global_load_async_to_lds_b128(
        (gv4i*)(esrc + idx4), (lv4i*)(et + j * 68 + kq), 0, 0);
#else
    *(float4*)(et + j * 68 + kq) = esrc[idx4];
#endif
  }
  // stage We transposed: WeT[n][k] = We[k][n]
  for (int idx = tid; idx < 64 * 64; idx += 256) {
    const int n = idx & 63, k = idx >> 6;
    wt[n * 68 + k] = We[k * 64 + n];
  }
  if (tid < 64) ae[tid] = a_edge[tid];
#if HAVE_ASYNC_LDS
  wait_async0();
#endif
  __syncthreads();

  const int l15 = lane & 15, lh = lane >> 4;
  float* eo = e_out + ebase;
  for (int nt = 0; nt < 4; ++nt) {             // column tiles (outer)
    const int n0 = nt << 4;
    v2f breg[16];
#pragma unroll
    for (int kk = 0; kk < 16; ++kk)            // hold B frags in 32 VGPRs
      breg[kk] = *(const v2f*)(wt + (n0 + l15) * 68 + (kk << 2) + lh * 2);
    const float aev = ae[n0 + l15];            // a_edge flat idx == column
    const int h = (n0 + l15) >> 3;

    for (int jt = wave; jt < 32; jt += 8) {    // 4 row tiles per wave
      const int j0 = jt << 4;
      v8f c = {0.f, 0.f, 0.f, 0.f, 0.f, 0.f, 0.f, 0.f};
#pragma unroll
      for (int kk = 0; kk < 16; ++kk) {        // K = 64 in steps of 4
        v2f a = *(const v2f*)(et + (j0 + l15) * 68 + (kk << 2) + lh * 2);
        c = wmma4(a, breg[kk], c);
      }
      // C layout: lane L, vgpr r -> row j0 + r + (L>>4)*8, col n0 + (L&15)
#pragma unroll
      for (int r = 0; r < 8; ++r) {
        const int j = j0 + r + lh * 8;
        const float v = c[r];
        __builtin_nontemporal_store(elu1(v), eo + (size_t)j * K_ + n0 + l15);
        float p = v * aev;                     // reduce over f (8 lanes)
        p += __shfl_xor(p, 1); p += __shfl_xor(p, 2); p += __shfl_xor(p, 4);
        if ((lane & 7) == 0)
          sedge[((size_t)(b * H_ + h) * N_ + j) * N_ + i] = p;
      }
    }
  }
}

// ======================================================================
// Kernel 3: masked leaky-relu softmax over last axis, IN-PLACE in sedge.
// ======================================================================
__global__ void softmax_kernel(const float* __restrict__ adj,
                               const float* __restrict__ ssrc,
                               const float* __restrict__ sdst,
                               float* __restrict__ att) {
  __shared__ float rmax[8], rsum[8];
  const int bh = blockIdx.x >> 9;      // b*8+h
  const int p = blockIdx.x & 511;
  const int b = bh >> 3;
  const int t = threadIdx.x, lane = t & 31, warp = t >> 5;
  float* row = att + ((size_t)bh * N_ + p) * N_;
  const float* ar = adj + (size_t)(b * N_ + p) * N_;
  const float sp = ssrc[(bh << 9) + p];
  const int q0 = t, q1 = t + 256;
  float v0 = lrelu(sp + sdst[(bh << 9) + q0] + row[q0]);
  float v1 = lrelu(sp + sdst[(bh << 9) + q1] + row[q1]);
  if (ar[q0] < 0.5f) v0 = -__builtin_inff();
  if (ar[q1] < 0.5f) v1 = -__builtin_inff();
  float mx = fmaxf(v0, v1);
#pragma unroll
  for (int m = 16; m >= 1; m >>= 1) mx = fmaxf(mx, __shfl_xor(mx, m));
  if (lane == 0) rmax[warp] = mx;
  __syncthreads();
  float gm = rmax[0];
#pragma unroll
  for (int w = 1; w < 8; ++w) gm = fmaxf(gm, rmax[w]);
  const float e0 = __expf(v0 - gm), e1 = __expf(v1 - gm);
  float s = e0 + e1;
#pragma unroll
  for (int m = 16; m >= 1; m >>= 1) s += __shfl_xor(s, m);
  if (lane == 0) rsum[warp] = s;
  __syncthreads();
  float gs = rsum[0];
#pragma unroll
  for (int w = 1; w < 8; ++w) gs += rsum[w];
  const float inv = 1.0f / gs;
  row[q0] = e0 * inv;
  row[q1] = e1 * inv;
}

// ======================================================================
// Kernel 4: out[b,i,h,f] = sum_j att[b,h,i,j]*xp[b,j,h,f]  (WMMA)
//           then + x + bias, elu.  grid = (B*H, 4 i-chunks), 256 thr.
// ======================================================================
__global__ void attend_kernel(const float* __restrict__ att,
                              const float* __restrict__ xp,
                              const float* __restrict__ x,
                              const float* __restrict__ bias,
                              float* __restrict__ out) {
  __shared__ float xphT[F_ * 516];     // xp head slice transposed [f][j]
  const int bh = blockIdx.x, b = bh >> 3, h = bh & 7;
  const int tid = threadIdx.x, lane = tid & 31, wave = tid >> 5;
  for (int idx = tid; idx < N_ * F_; idx += 256) {
    const int j = idx >> 3, f = idx & 7;
    xphT[f * 516 + j] = xp[(size_t)(b * N_ + j) * K_ + h * F_ + f];
  }
  __syncthreads();

  const int l15 = lane & 15, lh = lane >> 4;
  const float fm = (l15 < 8) ? 1.f : 0.f;          // zero invalid B columns
  const int fs = l15 & 7;
  const int i0 = (blockIdx.y * 8 + wave) << 4;     // one 16-row tile per wave
  const float* arow = att + ((size_t)bh * N_ + i0 + l15) * N_;
  v8f c = {0.f, 0.f, 0.f, 0.f, 0.f, 0.f, 0.f, 0.f};
#pragma unroll 8
  for (int kk = 0; kk < 128; ++kk) {               // K = 512 in steps of 4
    v2f a = *(const v2f*)(arow + (kk << 2) + lh * 2);
    v2f bb = *(const v2f*)(xphT + fs * 516 + (kk << 2) + lh * 2);
    bb.x *= fm; bb.y *= fm;                        // select, keeps EXEC uniform
    c = wmma4(a, bb, c);
  }
  if (l15 < 8) {
    const int col = h * F_ + l15;
    const float bv = bias[col];
#pragma unroll
    for (int r = 0; r < 8; ++r) {
      const int i = i0 + r + lh * 8;
      const size_t o = (size_t)(b * N_ + i) * K_ + col;
      out[o] = elu1(c[r] + x[o] + bv);
    }
  }
}

// ======================================================================
extern "C" void kernel_launch(void* const* d_in, const int* in_sizes, int n_in,
                              void* d_out, int out_size, void* d_ws, size_t ws_size,
                              hipStream_t stream) {
  const float* adj    = (const float*)d_in[0];
  const float* x      = (const float*)d_in[1];
  const float* e      = (const float*)d_in[2];
  const float* W      = (const float*)d_in[3];
  const float* We     = (const float*)d_in[4];
  const float* a_src  = (const float*)d_in[5];
  const float* a_dst  = (const float*)d_in[6];
  const float* a_edge = (const float*)d_in[7];
  const float* bias   = (const float*)d_in[8];

  float* out  = (float*)d_out;                  // B*N*K = 131072 floats
  float* eout = out + (size_t)B_ * N_ * K_;     // then B*N*N*K e_out

  float* ws    = (float*)d_ws;
  float* xp    = ws;                            // 131072
  float* ssrc  = xp + (size_t)B_ * N_ * K_;     // 16384
  float* sdst  = ssrc + (size_t)B_ * H_ * N_;   // 16384
  float* sedge = sdst + (size_t)B_ * H_ * N_;   // 8388608 (att in-place)

  proj_nodes_kernel<<<dim3(B_ * N_), dim3(64), 0, stream>>>(
      x, W, a_src, a_dst, xp, ssrc, sdst);

  const size_t smemB = (512 * 68 + 64 * 68 + 64) * sizeof(float);
  edge_kernel<<<dim3(B_ * N_), dim3(256), smemB, stream>>>(
      e, We, a_edge, eout, sedge);

  softmax_kernel<<<dim3(B_ * H_ * N_), dim3(256), 0, stream>>>(
      adj, ssrc, sdst, sedge);

  attend_kernel<<<dim3(B_ * H_, 4), dim3(256), 0, stream>>>(
      sedge, xp, x, bias, out);
}